// modelWithNMS_35845797052443
// MI455X (gfx1250) — compile-verified
//
#include <hip/hip_runtime.h>
#include <hip/hip_bf16.h>

// ---------------------------------------------------------------------------
// Batched NMS post-processing for MI455X (gfx1250, wave32, WGP, 320KB LDS)
//   B=8 images, A=25200 anchors, C=80 classes, TOPK=1000
// Pipeline:
//   K1: per-image fused class-max + mask + full top-k via 256KB-LDS bitonic
//   K2: per-image NMS:
//       - survivor boxes gathered into LDS via global_load_async_to_lds_b128
//         (ASYNCcnt-tracked, drained with s_wait_asynccnt)
//       - WMMA f32 16x16x4 pairwise area-sum tiles + divide-free IoU test
//       - ballot-packed 1024x1024 suppression bitmask in LDS (128KB)
//       - exact sequential keep-scan, masked output assembly
// ---------------------------------------------------------------------------

constexpr int   Bn          = 8;
constexpr int   An          = 25200;
constexpr int   Cn          = 80;
constexpr int   TOPK        = 1000;
constexpr int   NPAD        = 32768;          // next pow2 >= An
constexpr float CONF_THR    = 0.5f;
constexpr float IOU_THR     = 0.6f;
constexpr float CLASS_OFF   = 4096.0f;
constexpr float EPSI        = 1e-7f;

typedef unsigned long long u64;
typedef __attribute__((ext_vector_type(2))) float v2f;
typedef __attribute__((ext_vector_type(8))) float v8f;

__device__ __forceinline__ unsigned orderF(float f) {
    unsigned u = __float_as_uint(f);
    return (u & 0x80000000u) ? ~u : (u | 0x80000000u);
}
__device__ __forceinline__ float unorderF(unsigned ou) {
    unsigned u = (ou & 0x80000000u) ? (ou ^ 0x80000000u) : ~ou;
    return __uint_as_float(u);
}

// ---------------------------------------------------------------------------
// Kernel 1: per image — class max over 80, confidence mask, full descending
// bitonic sort of 32768 packed keys held entirely in 256KB dynamic LDS.
// Key = (orderedFloat(maskedConf) << 32) | (0xFFFFFFFF - anchorIdx)
//   -> descending sort == top_k with lowest-index tie-break (matches lax.top_k)
// ---------------------------------------------------------------------------
__global__ __launch_bounds__(1024)
void topk_kernel(const float* __restrict__ scores,
                 float* __restrict__ topv, int* __restrict__ topi) {
    extern __shared__ u64 keys[];                     // 32768 * 8B = 256KB LDS
    const int b   = blockIdx.x;
    const int tid = threadIdx.x;

    for (int t = tid; t < NPAD; t += 1024) {
        u64 key = 0ull;                               // padding: smallest key
        if (t < An) {
            const float4* sp = (const float4*)(scores + ((size_t)b * An + t) * Cn);
            if (t + 1024 < An)
                __builtin_prefetch(sp + 20 * 1024, 0, 0);  // -> global_prefetch_b8
            float mv = -1e30f;
#pragma unroll
            for (int g = 0; g < 20; ++g) {
                float4 s = sp[g];
                mv = fmaxf(mv, fmaxf(fmaxf(s.x, s.y), fmaxf(s.z, s.w)));
            }
            float masked = (mv >= CONF_THR) ? mv : -1.0f;
            key = ((u64)orderF(masked) << 32) | (u64)(0xFFFFFFFFu - (unsigned)t);
        }
        keys[t] = key;
    }
    __syncthreads();

    // Bitonic sort, descending. 120 stages, 16 compare-exchanges/thread/stage.
    for (int k = 2; k <= NPAD; k <<= 1) {
        for (int j = k >> 1; j > 0; j >>= 1) {
            for (int t = tid; t < NPAD; t += 1024) {
                int ixj = t ^ j;
                if (ixj > t) {
                    u64 a = keys[t];
                    u64 c = keys[ixj];
                    bool desc = ((t & k) == 0);
                    if (desc ? (a < c) : (a > c)) { keys[t] = c; keys[ixj] = a; }
                }
            }
            __syncthreads();
        }
    }

    if (tid < TOPK) {
        u64 key = keys[tid];
        topv[b * TOPK + tid] = unorderF((unsigned)(key >> 32));
        topi[b * TOPK + tid] = (int)(0xFFFFFFFFu - (unsigned)(key & 0xFFFFFFFFull));
    }
}

// ---------------------------------------------------------------------------
// Kernel 2: per image NMS on the 1000 survivors.
// ---------------------------------------------------------------------------
__global__ __launch_bounds__(512)
void nms_kernel(const float* __restrict__ boxes,
                const float* __restrict__ scores,
                const float* __restrict__ topv,
                const int*  __restrict__ topi,
                float* __restrict__ out) {
    extern __shared__ float4 smem_f4[];
    char* smem = (char*)smem_f4;
    float4*   boxS    = (float4*)smem;                          // 1024*16 = 16384
    float*    offS    = (float*)(smem + 16384);                 // 1024*4
    float*    areaS   = (float*)(smem + 16384 + 4096);          // 1024*4
    float*    confS   = (float*)(smem + 16384 + 8192);          // 1024*4
    float*    clsS    = (float*)(smem + 16384 + 12288);         // 1024*4
    unsigned* rowbits = (unsigned*)(smem + 32768);              // 1024*32*4 = 131072
    unsigned* keepS   = (unsigned*)(smem + 32768 + 131072);     // 32*4

    const int b   = blockIdx.x;
    const int tid = threadIdx.x;

    // ---------------- Phase A: gather survivors into LDS -------------------
    // Box coords: async DMA gather, global -> LDS, tracked by ASYNCcnt.
    for (int t = tid; t < TOPK; t += 512) {
        int gi = topi[b * TOPK + t];
        const float* gp = boxes + ((size_t)b * An + gi) * 4;
        unsigned ldsAddr = (unsigned)(uintptr_t)(void*)&boxS[t];
        asm volatile("global_load_async_to_lds_b128 %0, %1, off"
                     :: "v"(ldsAddr), "v"(gp) : "memory");
    }
    // Class argmax / offsets / conf for survivors (VALU path, overlaps DMA).
    for (int t = tid; t < 1024; t += 512) {
        float v = -1.0f, off = 0.0f, clsf = 0.0f;
        if (t < TOPK) {
            int gi = topi[b * TOPK + t];
            v = topv[b * TOPK + t];
            const float* sp = scores + ((size_t)b * An + gi) * Cn;
            float mv = -1e30f; int mi = 0;
            for (int c = 0; c < Cn; ++c) {
                float s = sp[c];
                if (s > mv) { mv = s; mi = c; }
            }
            off  = (float)mi * CLASS_OFF;
            clsf = (float)mi;
        } else {
            // padding rows: far-apart degenerate boxes, never kept
            float far = (float)t * 1.0e7f;
            boxS[t] = make_float4(far, far, far, far);
        }
        offS[t]  = off;
        confS[t] = v;
        clsS[t]  = clsf;
    }
    for (int w = tid; w < 1024 * 32; w += 512) rowbits[w] = 0u;
    if (tid < 32) keepS[tid] = 0u;
    asm volatile("s_wait_asynccnt 0" ::: "memory");   // DMA gather complete
    __syncthreads();

    // areas (need async-gathered boxes)
    for (int t = tid; t < 1024; t += 512) {
        float4 bx = boxS[t];
        areaS[t] = (bx.z - bx.x) * (bx.w - bx.y);
    }
    for (int t = tid; t < TOPK; t += 512)
        if (confS[t] >= CONF_THR)
            atomicOr(&keepS[t >> 5], 1u << (t & 31));
    __syncthreads();

    // ---------------- Phase B: suppression bitmask via WMMA tiles ----------
    const int lane = tid & 31;
    const int wave = tid >> 5;
    const int n    = lane & 15;
    for (int Iblk = wave; Iblk < 64; Iblk += 16) {           // wave-uniform
        // A(16x4): row m = [area_i, 1, 0, 0]
        // layout: lanes 0-15 carry K=0/1 in v0/v1, lanes 16-31 K=2/3 (zeros)
        float ax = 0.f, ay = 0.f;
        if (lane < 16) { ax = areaS[Iblk * 16 + lane]; ay = 1.0f; }
        v2f Am = { ax, ay };
        // Hoist this lane's 8 row boxes (rows r or r+8 of the D layout).
        const int rowbase = Iblk * 16 + ((lane < 16) ? 0 : 8);
        float4 bi[8]; float oi[8];
#pragma unroll
        for (int r = 0; r < 8; ++r) {
            bi[r] = boxS[rowbase + r];
            oi[r] = offS[rowbase + r];
        }
        for (int J = 0; J < 64; ++J) {
            const int j = J * 16 + n;
            float4 bj = boxS[j];
            float  oj = offS[j];
            // B(4x16): col n = [1, area_j, 0, 0]
            float bxv = 0.f, byv = 0.f;
            if (lane < 16) { bxv = 1.0f; byv = areaS[J * 16 + lane]; }
            v2f Bm = { bxv, byv };
            v8f Cz = {};
            v8f D = __builtin_amdgcn_wmma_f32_16x16x4_f32(
                false, Am, false, Bm, (short)0, Cz, false, false);
#pragma unroll
            for (int r = 0; r < 8; ++r) {
                const int i = Iblk * 16 + ((lane < 16) ? r : (r + 8));
                float xx1 = fmaxf(bi[r].x + oi[r], bj.x + oj);
                float yy1 = fmaxf(bi[r].y + oi[r], bj.y + oj);
                float xx2 = fminf(bi[r].z + oi[r], bj.z + oj);
                float yy2 = fminf(bi[r].w + oi[r], bj.w + oj);
                float inter = fmaxf(xx2 - xx1, 0.f) * fmaxf(yy2 - yy1, 0.f);
                // iou > thr  <=>  inter > thr*(aSum - inter + eps); denom > 0
                float rhs = IOU_THR * (D[r] - inter + EPSI);
                bool  sup = (inter > rhs) && (j > i) && (i < TOPK) && (j < TOPK);
                unsigned mask = (unsigned)__ballot(sup);     // wave32: 32 bits
                if (lane == 0) {
                    unsigned sh = (unsigned)(J & 1) * 16u;
                    atomicOr(&rowbits[(Iblk * 16 + r)     * 32 + (J >> 1)],
                             (mask & 0xFFFFu) << sh);
                    atomicOr(&rowbits[(Iblk * 16 + 8 + r) * 32 + (J >> 1)],
                             (mask >> 16) << sh);
                }
            }
        }
    }
    __syncthreads();

    // ---------------- Phase C: sequential keep-scan (matches lax.scan) -----
    if (tid < 32) {
        for (int i = 0; i < TOPK; ++i) {
            unsigned kw = keepS[i >> 5];
            if (kw & (1u << (i & 31)))
                keepS[tid] &= ~rowbits[i * 32 + tid];
        }
    }
    __syncthreads();

    // ---------------- Phase D: output assembly ----------------
    for (int t = tid; t < TOPK; t += 512) {
        float fk = (float)((keepS[t >> 5] >> (t & 31)) & 1u);
        float4 bx = boxS[t];
        float* o = out + ((size_t)b * TOPK + t) * 8;
        o[0] = confS[t] * fk;
        o[1] = bx.x * fk;
        o[2] = bx.y * fk;
        o[3] = bx.z * fk;
        o[4] = bx.w * fk;
        o[5] = (float)b * fk;
        o[6] = clsS[t] * fk;
        o[7] = 0.0f;
    }
}

// ---------------------------------------------------------------------------
extern "C" void kernel_launch(void* const* d_in, const int* in_sizes, int n_in,
                              void* d_out, int out_size, void* d_ws, size_t ws_size,
                              hipStream_t stream) {
    (void)in_sizes; (void)n_in; (void)out_size; (void)ws_size;
    const float* boxes  = (const float*)d_in[0];   // (8, 25200, 4)
    const float* scores = (const float*)d_in[1];   // (8, 25200, 80)
    float* topv = (float*)d_ws;                                    // 8*1000 f32
    int*   topi = (int*)((char*)d_ws + Bn * TOPK * sizeof(float)); // 8*1000 i32
    float* outF = (float*)d_out;                   // (8, 1000, 8)

    const int smem1 = NPAD * (int)sizeof(u64);           // 262144 B
    const int smem2 = 32768 + 131072 + 128;              // 163968 B
    (void)hipFuncSetAttribute((const void*)topk_kernel,
                              hipFuncAttributeMaxDynamicSharedMemorySize, smem1);
    (void)hipFuncSetAttribute((const void*)nms_kernel,
                              hipFuncAttributeMaxDynamicSharedMemorySize, smem2);

    topk_kernel<<<Bn, 1024, smem1, stream>>>(scores, topv, topi);
    nms_kernel<<<Bn, 512, smem2, stream>>>(boxes, scores, topv, topi, outF);
}